// DCTLayer_55104430408209
// MI455X (gfx1250) — compile-verified
//
#include <hip/hip_runtime.h>

typedef __attribute__((ext_vector_type(2))) float v2f;
typedef __attribute__((ext_vector_type(8))) float v8f;

#define IMG_H 256
#define IMG_W 256
#define LDS_STRIDE 18           // 16 cols + 2 pad: conflict-free & keeps v2f 8B-aligned
#define WAVES_PER_BLOCK 8
#define MAX_BLOCKS 2048

// Per-wave 16x16 fp32 tile DCT: out = B16 * X * B16^T, B16 = blockdiag(D,D,D,D).
// Stage 1: P = B16 * X           (A = const B16, B = X loaded in B-operand layout)
// Stage 2: out = P * B16^T       (A = P via LDS transpose, B = same const regs)
__global__ __launch_bounds__(256) void dct4x4_wmma_kernel(
    const float* __restrict__ x, const float* __restrict__ Dm,
    float* __restrict__ out, int numTiles) {
  __shared__ __align__(16) float lds[WAVES_PER_BLOCK][16 * LDS_STRIDE];

  const int lane = threadIdx.x & 31;
  const int wv   = threadIdx.x >> 5;
  const int col  = lane & 15;      // N (and M for A-operand reads)
  const int hh   = lane >> 4;      // which K sub-pair this lane holds
  const int tb   = hh * 2;

  // The only two DCT-matrix entries this lane ever needs:
  const float dv0 = Dm[(col & 3) * 4 + tb];
  const float dv1 = Dm[(col & 3) * 4 + tb + 1];
  const int blk = col >> 2;

  // Zero-masked block-diagonal operand per k-step (serves as stage-1 A and stage-2 B)
  v2f cst[4];
#pragma unroll
  for (int kk = 0; kk < 4; ++kk) {
    cst[kk][0] = (blk == kk) ? dv0 : 0.0f;
    cst[kk][1] = (blk == kk) ? dv1 : 0.0f;
  }

  float* tl = &lds[wv][0];
  const int waveId = blockIdx.x * WAVES_PER_BLOCK + wv;
  const int nWaves = gridDim.x * WAVES_PER_BLOCK;

  for (int t = waveId; t < numTiles; t += nWaves) {
    // 256 tiles per 256x256 image, 16 tiles per row
    const int img  = t >> 8;
    const int trow = ((t >> 4) & 15) << 4;
    const int tcol = (t & 15) << 4;
    const size_t base = (size_t)img * (IMG_H * IMG_W) + (size_t)trow * IMG_W + tcol;

    // ---- coalesced load of X straight into WMMA B-operand layout ----
    v2f bx[4];
#pragma unroll
    for (int kk = 0; kk < 4; ++kk) {
      const int r0 = 4 * kk + tb;
      bx[kk][0] = x[base + (size_t)r0 * IMG_W + col];
      bx[kk][1] = x[base + (size_t)(r0 + 1) * IMG_W + col];
    }

    // ---- stage 1: P = B16 * X ----
    v8f p = {0.f, 0.f, 0.f, 0.f, 0.f, 0.f, 0.f, 0.f};
#pragma unroll
    for (int kk = 0; kk < 4; ++kk)
      p = __builtin_amdgcn_wmma_f32_16x16x4_f32(false, cst[kk], false, bx[kk],
                                                (short)0, p, false, false);

    // ---- C-layout -> LDS (row-major, padded); intra-wave DS is in-order ----
#pragma unroll
    for (int r = 0; r < 8; ++r)
      tl[(r + 8 * hh) * LDS_STRIDE + col] = p[r];

    // ---- stage 2: out = P * B16^T (A from LDS in A-operand layout) ----
    v8f o = {0.f, 0.f, 0.f, 0.f, 0.f, 0.f, 0.f, 0.f};
#pragma unroll
    for (int kk = 0; kk < 4; ++kk) {
      v2f a = *(const v2f*)&tl[col * LDS_STRIDE + 4 * kk + tb];
      o = __builtin_amdgcn_wmma_f32_16x16x4_f32(false, a, false, cst[kk],
                                                (short)0, o, false, false);
    }

    // ---- coalesced store: result is in C layout == row-major rows of the tile ----
#pragma unroll
    for (int r = 0; r < 8; ++r)
      out[base + (size_t)(r + 8 * hh) * IMG_W + col] = o[r];
  }
}

extern "C" void kernel_launch(void* const* d_in, const int* in_sizes, int n_in,
                              void* d_out, int out_size, void* d_ws, size_t ws_size,
                              hipStream_t stream) {
  const float* x  = (const float*)d_in[0];
  const float* Dm = (const float*)d_in[1];
  float* out = (float*)d_out;

  const int numTiles = in_sizes[0] / 256;   // 16x16-element tiles
  int grid = (numTiles + WAVES_PER_BLOCK - 1) / WAVES_PER_BLOCK;
  if (grid > MAX_BLOCKS) grid = MAX_BLOCKS;
  if (grid < 1) grid = 1;

  dct4x4_wmma_kernel<<<grid, 32 * WAVES_PER_BLOCK, 0, stream>>>(x, Dm, out, numTiles);
}